// PaiNet_35064113005155
// MI455X (gfx1250) — compile-verified
//
#include <hip/hip_runtime.h>
#include <hip/hip_bf16.h>

#define BATCH   8
#define KNN     20
#define NKERN   9
#define KDQ     5
#define NGROUP  4
#define CHUNKR  81920
#define BN_SC   0.9999950000374997f
#define CDIV(a,b) (((a)+(b)-1)/(b))

typedef float v2f __attribute__((ext_vector_type(2)));
typedef float v8f __attribute__((ext_vector_type(8)));

__device__ __forceinline__ float gelu_f(float x) {
  return 0.5f * x * (1.0f + erff(x * 0.7071067811865475f));
}

// ---------------------------------------------------------------------------
// Generic strided-batch GEMM with fused bias/scale/GELU using WMMA f32 16x16x4.
// C[b][m][n] = act( scale * ( sum_k A[b][m][k] * B[b][k][n] + bias ) )
// transB: B stored as (N x K) row-major instead of (K x N).
// One wave (32 threads) computes one 16(M) x 32(N) C tile (two accumulators).
// All operand loads are branchless: indices are clamped in-bounds and the
// loaded value is selected to zero outside the tile, so EXEC never changes
// and the K-loop is pure load + v_cndmask + v_wmma.
// ---------------------------------------------------------------------------
__global__ void k_gemm(const float* __restrict__ A, const float* __restrict__ Bm,
                       const float* __restrict__ bias, float* __restrict__ C,
                       int M, int N, int Kd, int lda, int ldb, int ldc,
                       long sA, long sB, long sC, int transB, int biasDim,
                       int act, float scale)
{
  int bz = blockIdx.z;
  const float* Ab = A  + (long)bz * sA;
  const float* Bb = Bm + (long)bz * sB;
  float*       Cb = C  + (long)bz * sC;
  int tile_n = blockIdx.x << 5;       // 32 output columns per wave
  int tile_m = blockIdx.y << 4;       // 16 output rows per wave
  int lane = threadIdx.x;
  int half = lane >> 4;               // 0: K pair {0,1}, 1: K pair {2,3}
  int l16  = lane & 15;

  int am  = tile_m + l16;             // A row for this lane
  int bn0 = tile_n + l16;             // B col, first 16-wide group
  int bn1 = bn0 + 16;                 // B col, second group
  bool aIn  = (am  < M);
  bool b0In = (bn0 < N);
  bool b1In = (bn1 < N);
  int amc  = aIn  ? am  : (M - 1);    // clamped (always legal) indices
  int bn0c = b0In ? bn0 : (N - 1);
  int bn1c = b1In ? bn1 : (N - 1);

  const float* Arow  = Ab + (long)amc * lda;
  const float* Br0   = Bb + (long)bn0c * ldb;   // used when transB
  const float* Br1   = Bb + (long)bn1c * ldb;

  v8f acc0 = {0.f,0.f,0.f,0.f,0.f,0.f,0.f,0.f};
  v8f acc1 = {0.f,0.f,0.f,0.f,0.f,0.f,0.f,0.f};

  int kMain = Kd & ~3;
  int ka = half * 2;                  // this lane's first K index per step
  if (transB) {
    for (int k0 = 0; k0 < kMain; k0 += 4, ka += 4) {
      float a0 = Arow[ka], a1 = Arow[ka + 1];
      float b00 = Br0[ka], b01 = Br0[ka + 1];
      float b10 = Br1[ka], b11 = Br1[ka + 1];
      v2f av = { aIn  ? a0  : 0.f, aIn  ? a1  : 0.f };
      v2f v0 = { b0In ? b00 : 0.f, b0In ? b01 : 0.f };
      v2f v1 = { b1In ? b10 : 0.f, b1In ? b11 : 0.f };
      acc0 = __builtin_amdgcn_wmma_f32_16x16x4_f32(false, av, false, v0,
                                                   (short)0, acc0, false, false);
      acc1 = __builtin_amdgcn_wmma_f32_16x16x4_f32(false, av, false, v1,
                                                   (short)0, acc1, false, false);
    }
  } else {
    for (int k0 = 0; k0 < kMain; k0 += 4, ka += 4) {
      float a0 = Arow[ka], a1 = Arow[ka + 1];
      const float* Bk0 = Bb + (long)ka * ldb;
      const float* Bk1 = Bk0 + ldb;
      float b00 = Bk0[bn0c], b01 = Bk1[bn0c];
      float b10 = Bk0[bn1c], b11 = Bk1[bn1c];
      v2f av = { aIn  ? a0  : 0.f, aIn  ? a1  : 0.f };
      v2f v0 = { b0In ? b00 : 0.f, b0In ? b01 : 0.f };
      v2f v1 = { b1In ? b10 : 0.f, b1In ? b11 : 0.f };
      acc0 = __builtin_amdgcn_wmma_f32_16x16x4_f32(false, av, false, v0,
                                                   (short)0, acc0, false, false);
      acc1 = __builtin_amdgcn_wmma_f32_16x16x4_f32(false, av, false, v1,
                                                   (short)0, acc1, false, false);
    }
  }
  if (kMain < Kd) {                   // masked K tail (one step)
    int kA = ka, kB = ka + 1;
    bool kv0 = kA < Kd, kv1 = kB < Kd;
    int kAc = kv0 ? kA : (Kd - 1);
    int kBc = kv1 ? kB : (Kd - 1);
    float a0 = Arow[kAc], a1 = Arow[kBc];
    float b00, b01, b10, b11;
    if (transB) {
      b00 = Br0[kAc]; b01 = Br0[kBc];
      b10 = Br1[kAc]; b11 = Br1[kBc];
    } else {
      const float* Bk0 = Bb + (long)kAc * ldb;
      const float* Bk1 = Bb + (long)kBc * ldb;
      b00 = Bk0[bn0c]; b01 = Bk1[bn0c];
      b10 = Bk0[bn1c]; b11 = Bk1[bn1c];
    }
    v2f av = { (aIn  && kv0) ? a0  : 0.f, (aIn  && kv1) ? a1  : 0.f };
    v2f v0 = { (b0In && kv0) ? b00 : 0.f, (b0In && kv1) ? b01 : 0.f };
    v2f v1 = { (b1In && kv0) ? b10 : 0.f, (b1In && kv1) ? b11 : 0.f };
    acc0 = __builtin_amdgcn_wmma_f32_16x16x4_f32(false, av, false, v0,
                                                 (short)0, acc0, false, false);
    acc1 = __builtin_amdgcn_wmma_f32_16x16x4_f32(false, av, false, v1,
                                                 (short)0, acc1, false, false);
  }

#pragma unroll
  for (int i = 0; i < 8; ++i) {
    int cm = tile_m + half * 8 + i;   // C/D layout: lanes<16 -> M=i, >=16 -> M=8+i
    if (cm < M) {
      if (b0In) {
        float v = acc0[i];
        if (bias) v += (biasDim == 0) ? bias[cm] : bias[bn0];
        v *= scale;
        if (act) v = gelu_f(v);
        Cb[(long)cm * ldc + bn0] = v;
      }
      if (b1In) {
        float v = acc1[i];
        if (bias) v += (biasDim == 0) ? bias[cm] : bias[bn1];
        v *= scale;
        if (act) v = gelu_f(v);
        Cb[(long)cm * ldc + bn1] = v;
      }
    }
  }
}

// ---------------------------------------------------------------------------
// (B,3,N) -> (B,N,3)
__global__ void k_xpose(const float* __restrict__ x, float* __restrict__ pts,
                        int Bn, int N) {
  int tid = blockIdx.x * blockDim.x + threadIdx.x;
  int total = Bn * N * 3;
  if (tid >= total) return;
  int b = tid / (N * 3);
  int rem = tid - b * N * 3;
  int n = rem / 3, d = rem - n * 3;
  pts[tid] = x[((long)b * 3 + d) * N + n];
}

// PK = ((A + A^T)/2) @ kernels   (3x9)
__global__ void k_pk(const float* __restrict__ A, const float* __restrict__ kern,
                     float* __restrict__ PK) {
  int t = threadIdx.x;
  if (t >= 27) return;
  int d = t / 9, m = t - d * 9;
  float v = 0.f;
  for (int e = 0; e < 3; ++e)
    v += 0.5f * (A[d * 3 + e] + A[e * 3 + d]) * kern[e * 9 + m];
  PK[d * 9 + m] = v;
}

// brute-force knn, K=20 smallest distances, ties keep earlier index
__global__ void k_knn(const float* __restrict__ pts, int* __restrict__ idx,
                      int Bn, int N) {
  int tid = blockIdx.x * blockDim.x + threadIdx.x;
  if (tid >= Bn * N) return;
  int b = tid / N, n = tid - b * N;
  const float* P = pts + (long)b * N * 3;
  float px = P[n*3], py = P[n*3+1], pz = P[n*3+2];
  float bd[KNN]; int bi[KNN];
  for (int i = 0; i < KNN; ++i) { bd[i] = 3.4e38f; bi[i] = 0; }
  for (int j = 0; j < N; ++j) {
    float dx = P[j*3] - px, dy = P[j*3+1] - py, dz = P[j*3+2] - pz;
    float d = dx*dx + dy*dy + dz*dz;
    if (d < bd[KNN-1]) {
      int p = KNN - 1;
      while (p > 0 && d < bd[p-1]) { bd[p] = bd[p-1]; bi[p] = bi[p-1]; --p; }
      bd[p] = d; bi[p] = j;
    }
  }
  int* o = idx + (long)tid * KNN;
  for (int i = 0; i < KNN; ++i) o[i] = bi[i];
}

// xf = [x_rep(3), x_rel(3), dist(1)] per (b,n,k)
__global__ void k_xf(const float* __restrict__ pts, const int* __restrict__ idx,
                     float* __restrict__ xf, int Bn, int N) {
  int r = blockIdx.x * blockDim.x + threadIdx.x;
  long R = (long)Bn * N * KNN;
  if (r >= R) return;
  int bn = r / KNN;
  int b = bn / N;
  const float* P = pts + (long)b * N * 3;
  int j  = idx[r];
  int j0 = idx[(long)bn * KNN];
  float rx = P[j0*3], ry = P[j0*3+1], rz = P[j0*3+2];
  float ax = P[j*3] - rx, ay = P[j*3+1] - ry, az = P[j*3+2] - rz;
  float dis = sqrtf(ax*ax + ay*ay + az*az + 1e-12f);
  float* o = xf + (long)r * 7;
  o[0]=rx; o[1]=ry; o[2]=rz; o[3]=ax; o[4]=ay; o[5]=az; o[6]=dis;
}

__global__ void k_minmax_part(const float* __restrict__ v, long n,
                              float* __restrict__ pmin, float* __restrict__ pmax) {
  __shared__ float smn[256], smx[256];
  float mn = 3.4e38f, mx = -3.4e38f;
  for (long i = (long)blockIdx.x * 256 + threadIdx.x; i < n; i += (long)gridDim.x * 256) {
    float x = v[i]; mn = fminf(mn, x); mx = fmaxf(mx, x);
  }
  smn[threadIdx.x] = mn; smx[threadIdx.x] = mx; __syncthreads();
  for (int s = 128; s > 0; s >>= 1) {
    if ((int)threadIdx.x < s) {
      smn[threadIdx.x] = fminf(smn[threadIdx.x], smn[threadIdx.x + s]);
      smx[threadIdx.x] = fmaxf(smx[threadIdx.x], smx[threadIdx.x + s]);
    }
    __syncthreads();
  }
  if (threadIdx.x == 0) { pmin[blockIdx.x] = smn[0]; pmax[blockIdx.x] = smx[0]; }
}

__global__ void k_minmax_fin(const float* __restrict__ pmin, const float* __restrict__ pmax,
                             int n, float* __restrict__ sout) {
  __shared__ float smn[256], smx[256];
  float mn = 3.4e38f, mx = -3.4e38f;
  for (int i = threadIdx.x; i < n; i += 256) {
    mn = fminf(mn, pmin[i]); mx = fmaxf(mx, pmax[i]);
  }
  smn[threadIdx.x] = mn; smx[threadIdx.x] = mx; __syncthreads();
  for (int s = 128; s > 0; s >>= 1) {
    if ((int)threadIdx.x < s) {
      smn[threadIdx.x] = fminf(smn[threadIdx.x], smn[threadIdx.x + s]);
      smx[threadIdx.x] = fmaxf(smx[threadIdx.x], smx[threadIdx.x + s]);
    }
    __syncthreads();
  }
  if (threadIdx.x == 0) sout[0] = smn[0] / (smx[0] - smn[0]);
}

// sc[j][rl] = sin(v_j), sc[32+j][rl] = cos(v_j), v_j = 2pi * sum_c (xf-s)*Bf[c][j]
__global__ void k_sincos(const float* __restrict__ xf, const float* __restrict__ Bf,
                         const float* __restrict__ sbuf, float* __restrict__ sc,
                         long r0, int Rc) {
  int rl = blockIdx.x * blockDim.x + threadIdx.x;
  int j = blockIdx.y;
  if (rl >= Rc) return;
  const float* xr = xf + (r0 + rl) * 7;
  float s = sbuf[0];
  float v = 0.f;
#pragma unroll
  for (int c = 0; c < 7; ++c) v += (xr[c] - s) * Bf[c * 32 + j];
  v *= 6.283185307179586f;
  sc[(long)j * Rc + rl]        = sinf(v);
  sc[(long)(32 + j) * Rc + rl] = cosf(v);
}

// perm[r][m] = x_rel . PK[:,m]  (+ one_pad at k==0,m==0)
__global__ void k_perm(const float* __restrict__ xf, const float* __restrict__ PK,
                       float* __restrict__ pm, long R) {
  long r = (long)blockIdx.x * blockDim.x + threadIdx.x;
  if (r >= R) return;
  const float* xr = xf + r * 7;
  float ax = xr[3], ay = xr[4], az = xr[5];
  int k = (int)(r % KNN);
  float* o = pm + r * NKERN;
#pragma unroll
  for (int m = 0; m < NKERN; ++m) {
    float p = ax * PK[m] + ay * PK[9 + m] + az * PK[18 + m];
    if (k == 0 && m == 0) p += 1.0f;
    o[m] = p;
  }
}

// topkmax (keep top-3 along K, relu) then normalize over K (in place)
__global__ void k_topk(float* __restrict__ pm, int BN) {
  int tid = blockIdx.x * blockDim.x + threadIdx.x;
  if (tid >= BN * NKERN) return;
  int bn = tid / NKERN, m = tid - bn * NKERN;
  float v[KNN];
  float t1 = -3.4e38f, t2 = -3.4e38f, t3 = -3.4e38f;
#pragma unroll
  for (int k = 0; k < KNN; ++k) {
    float x = pm[((long)bn * KNN + k) * NKERN + m];
    v[k] = x;
    if (x > t1)      { t3 = t2; t2 = t1; t1 = x; }
    else if (x > t2) { t3 = t2; t2 = x; }
    else if (x > t3) { t3 = x; }
  }
  float sum = 0.f;
#pragma unroll
  for (int k = 0; k < KNN; ++k) {
    float x = (v[k] >= t3) ? v[k] : 0.f;
    x = fmaxf(x, 0.f);
    v[k] = x; sum += x;
  }
  float inv = 1.f / (sum + 1e-6f);
#pragma unroll
  for (int k = 0; k < KNN; ++k)
    pm[((long)bn * KNN + k) * NKERN + m] = v[k] * inv;
}

// agg[c][(bn)*9+m] = sum_k feats_shuffled[bn,k,c] * pm[bn,k,m], feats built on the fly
__global__ void k_agg(const float* __restrict__ feat, const float* __restrict__ ff,
                      const int* __restrict__ idx, const float* __restrict__ pm,
                      float* __restrict__ agg, int Bn, int N, int C, int icx, int nf) {
  long tid = (long)blockIdx.x * blockDim.x + threadIdx.x;
  long total = (long)Bn * N * nf;
  if (tid >= total) return;
  int bn = (int)(tid / nf);
  int c  = (int)(tid - (long)bn * nf);
  int b  = bn / N;
  int H = C + icx;
  int g = c % NGROUP;
  int c_src = g * (nf / NGROUP) + c / NGROUP;     // inverse channel shuffle
  int isdiff = (c_src >= H) ? 1 : 0;
  int c0 = isdiff ? (c_src - H) : c_src;
  long R = (long)Bn * N * KNN;
  long rbase = (long)bn * KNN;
  float rep;
  if (c0 < C) rep = feat[((long)b * C + c0) * N + idx[rbase]];
  else        rep = ff[(long)(c0 - C) * R + rbase];
  float acc[NKERN];
#pragma unroll
  for (int m = 0; m < NKERN; ++m) acc[m] = 0.f;
  for (int k = 0; k < KNN; ++k) {
    float vv;
    if (isdiff) {
      float cv;
      if (c0 < C) cv = feat[((long)b * C + c0) * N + idx[rbase + k]];
      else        cv = ff[(long)(c0 - C) * R + rbase + k];
      vv = cv - rep;
    } else {
      vv = rep;
    }
    const float* pw = pm + (rbase + k) * NKERN;
#pragma unroll
    for (int m = 0; m < NKERN; ++m) acc[m] += vv * pw[m];
  }
  long J = (long)Bn * N * NKERN;
  float* o = agg + (long)c * J + (long)bn * NKERN;
#pragma unroll
  for (int m = 0; m < NKERN; ++m) o[m] = acc[m];
}

// out = gelu( (gelu(max_m conv) + res) * BN_S )
__global__ void k_combine(const float* __restrict__ conv, const float* __restrict__ res,
                          float* __restrict__ out, int Bn, int N, int O) {
  long tid = (long)blockIdx.x * blockDim.x + threadIdx.x;
  long total = (long)Bn * O * N;
  if (tid >= total) return;
  int b = (int)(tid / ((long)O * N));
  long rem = tid - (long)b * O * N;
  int o = (int)(rem / N);
  int n = (int)(rem - (long)o * N);
  long bn = (long)b * N + n;
  long J = (long)Bn * N * NKERN;
  const float* p = conv + (long)o * J + bn * NKERN;
  float mx = p[0];
#pragma unroll
  for (int m = 1; m < NKERN; ++m) mx = fmaxf(mx, p[m]);
  float v = gelu_f(mx);
  v = (v + res[((long)b * O + o) * N + n]) * BN_SC;
  out[((long)b * O + o) * N + n] = gelu_f(v);
}

// furthest point sampling: one block per batch
__global__ void k_fps(const float* __restrict__ pts, float* __restrict__ dist,
                      int* __restrict__ fidx, int N, int S) {
  int b = blockIdx.x;
  int t = threadIdx.x;
  __shared__ float sv[256];
  __shared__ int   si[256];
  __shared__ int   s_last;
  const float* P = pts + (long)b * N * 3;
  float* D = dist + (long)b * N;
  for (int n = t; n < N; n += 256) D[n] = 1e10f;
  if (t == 0) s_last = 0;
  __syncthreads();
  for (int i = 0; i < S; ++i) {
    int last = s_last;
    if (t == 0) fidx[b * S + i] = last;
    float lx = P[last*3], ly = P[last*3+1], lz = P[last*3+2];
    float bd = -1.f; int bi = 0x7fffffff;
    for (int n = t; n < N; n += 256) {
      float dx = P[n*3]-lx, dy = P[n*3+1]-ly, dz = P[n*3+2]-lz;
      float d = dx*dx + dy*dy + dz*dz;
      float gmin = fminf(D[n], d);
      D[n] = gmin;
      if (gmin > bd) { bd = gmin; bi = n; }
    }
    sv[t] = bd; si[t] = bi; __syncthreads();
    for (int s = 128; s > 0; s >>= 1) {
      if (t < s) {
        if (sv[t+s] > sv[t] || (sv[t+s] == sv[t] && si[t+s] < si[t])) {
          sv[t] = sv[t+s]; si[t] = si[t+s];
        }
      }
      __syncthreads();
    }
    if (t == 0) s_last = si[0];
    __syncthreads();
  }
}

// ball query: first KD indices (ascending) within radius, pad with first
__global__ void k_ball(const float* __restrict__ pts, const int* __restrict__ fidx,
                       int* __restrict__ nidx, int Bn, int N, int S, float r2) {
  int tid = blockIdx.x * blockDim.x + threadIdx.x;
  if (tid >= Bn * S) return;
  int b = tid / S, s = tid - b * S;
  const float* P = pts + (long)b * N * 3;
  int c = fidx[b * S + s];
  float cx = P[c*3], cy = P[c*3+1], cz = P[c*3+2];
  int* o = nidx + (long)tid * KDQ;
  int cnt = 0, first = N - 1;
  for (int n = 0; n < N && cnt < KDQ; ++n) {
    float dx = P[n*3]-cx, dy = P[n*3+1]-cy, dz = P[n*3+2]-cz;
    float d = dx*dx + dy*dy + dz*dz;
    if (d <= r2) { if (cnt == 0) first = n; o[cnt++] = n; }
  }
  for (; cnt < KDQ; ++cnt) o[cnt] = first;
}

__global__ void k_ptsmax(const float* __restrict__ pts, const int* __restrict__ nidx,
                         float* __restrict__ ptsOut, int Bn, int N, int S) {
  int tid = blockIdx.x * blockDim.x + threadIdx.x;
  if (tid >= Bn * S) return;
  int b = tid / S, s = tid - b * S;
  const int* nd = nidx + (long)tid * KDQ;
  float mx0 = -3.4e38f, mx1 = -3.4e38f, mx2 = -3.4e38f;
  for (int k = 0; k < KDQ; ++k) {
    const float* p = pts + ((long)b * N + nd[k]) * 3;
    mx0 = fmaxf(mx0, p[0]); mx1 = fmaxf(mx1, p[1]); mx2 = fmaxf(mx2, p[2]);
  }
  float* o = ptsOut + ((long)b * S + s) * 3;
  o[0] = mx0; o[1] = mx1; o[2] = mx2;
}

__global__ void k_fgather(const float* __restrict__ feat, const int* __restrict__ nidx,
                          float* __restrict__ fg, int Bn, int N, int C, int S) {
  long tid = (long)blockIdx.x * blockDim.x + threadIdx.x;
  long total = (long)Bn * C * S;
  if (tid >= total) return;
  int b = (int)(tid / ((long)C * S));
  long rem = tid - (long)b * C * S;
  int c = (int)(rem / S);
  int s = (int)(rem - (long)c * S);
  const int* nd = nidx + ((long)b * S + s) * KDQ;
  const float* f = feat + ((long)b * C + c) * N;
  float* o = fg + (((long)b * C + c) * S + s) * KDQ;
#pragma unroll
  for (int k = 0; k < KDQ; ++k) o[k] = f[nd[k]];
}

__global__ void k_poolmax(const float* __restrict__ fgo, float* __restrict__ featOut,
                          int Bn, int C, int S) {
  long tid = (long)blockIdx.x * blockDim.x + threadIdx.x;
  long total = (long)Bn * C * S;
  if (tid >= total) return;
  const float* p = fgo + tid * KDQ;
  float mx = p[0];
#pragma unroll
  for (int k = 1; k < KDQ; ++k) mx = fmaxf(mx, p[k]);
  featOut[tid] = mx;
}

// h[b] = concat(max_n z[b,:,n], mean_n z[b,:,n]); z is (B,1024,64)
__global__ void k_head(const float* __restrict__ z, float* __restrict__ h, int Bn) {
  int tid = blockIdx.x * blockDim.x + threadIdx.x;
  if (tid >= Bn * 1024) return;
  int b = tid / 1024, c = tid - b * 1024;
  const float* p = z + ((long)b * 1024 + c) * 64;
  float mx = p[0], sm = 0.f;
  for (int n = 0; n < 64; ++n) { mx = fmaxf(mx, p[n]); sm += p[n]; }
  h[(long)b * 2048 + c]        = mx;
  h[(long)b * 2048 + 1024 + c] = sm * (1.f / 64.f);
}

// ---------------------------------------------------------------------------
struct ConvP { const float *A, *Bf, *mlpW, *mlpb, *convW, *convb, *outW; };

extern "C" void kernel_launch(void* const* d_in, const int* in_sizes, int n_in,
                              void* d_out, int out_size, void* d_ws, size_t ws_size,
                              hipStream_t stream) {
  auto F = [&](int i) { return (const float*)d_in[i]; };

  // Input ordering: source-dict insertion order (x first) vs jax-flatten
  // alphabetical (kernels first, groups alphabetical).
  bool alpha = (in_sizes[0] == 27);
  const int base[5] = {1, 8, 15, 22, 29};
  ConvP cv[5];
  const float *x, *kern, *pw[4];
  const float *conv5W, *lin1W, *lin2W, *lin2b, *lin3W, *lin3b;
  if (!alpha) {
    x = F(0); kern = F(46);
    for (int i = 0; i < 5; ++i) {
      int b0 = base[i];
      cv[i] = { F(b0+0), F(b0+1), F(b0+2), F(b0+3), F(b0+4), F(b0+5), F(b0+6) };
    }
    pw[0]=F(36); pw[1]=F(37); pw[2]=F(38); pw[3]=F(39);
    conv5W=F(40); lin1W=F(41); lin2W=F(42); lin2b=F(43); lin3W=F(44); lin3b=F(45);
  } else {
    kern = F(0); x = F(46);
    for (int i = 0; i < 5; ++i) {
      int b0 = base[i];   // alpha order within group: A,Bf,convW,convb,mlpW,mlpb,outW
      cv[i] = { F(b0+0), F(b0+1), F(b0+4), F(b0+5), F(b0+2), F(b0+3), F(b0+6) };
    }
    conv5W=F(36); lin1W=F(37); lin2W=F(38); lin2b=F(39); lin3W=F(40); lin3b=F(41);
    pw[0]=F(42); pw[1]=F(43); pw[2]=F(44); pw[3]=F(45);
  }

  // ---- workspace layout (floats) ----
  size_t off = 0;
  auto grab = [&](size_t n) { size_t r = off; off += n; return r; };
  const size_t F_PTS_A = grab(49152);
  const size_t F_PTS_B = grab(49152);
  const size_t F_FEAT_A = grab(1048576);
  const size_t F_FEAT_B = grab(1048576);
  const size_t F_IDX  = grab(327680);       // int
  const size_t F_XF   = grab(2293760);
  const size_t F_PM   = grab(2949120);
  const size_t F_FF   = grab(2621440);
  const size_t F_BIG  = grab(9437184);      // sincos chunk / conv output / pool scratch
  const size_t F_AGG  = grab(7077888);      // agg / fps-ball scratch
  const size_t F_RES  = grab(1048576);
  const size_t F_PK   = grab(32);
  const size_t F_SB   = grab(16);
  const size_t F_PMIN = grab(512);
  const size_t F_PMAX = grab(512);
  const size_t F_H    = grab(16384);
  const size_t F_H1   = grab(4096);
  const size_t F_H2   = grab(2048);
  if (ws_size < off * sizeof(float)) return;

  float* W = (float*)d_ws;
  float* ptsA = W + F_PTS_A;  float* ptsB = W + F_PTS_B;
  float* featBufs[2] = { W + F_FEAT_A, W + F_FEAT_B };
  int*   idxb = (int*)(W + F_IDX);
  float* xfb  = W + F_XF;
  float* pmb  = W + F_PM;
  float* ffb  = W + F_FF;
  float* bigb = W + F_BIG;                  // sincos / convout
  float* aggb = W + F_AGG;
  float* resb = W + F_RES;
  float* pkb  = W + F_PK;
  float* sb   = W + F_SB;
  float* pmin = W + F_PMIN;
  float* pmax = W + F_PMAX;
  float* hb   = W + F_H;
  float* h1b  = W + F_H1;
  float* h2b  = W + F_H2;
  // unions (safe: dead regions at time of use)
  float* fgb   = bigb;                      // pool gather   (<= 1,310,720)
  float* fgob  = bigb + 1310720;            // pool gemm out (<= 1,310,720)
  float* distb = aggb;                      // fps dist (16384)
  int*   fidxb = (int*)(aggb + 16384);      // fps idx  (4096)
  int*   nidxb = (int*)(aggb + 16384 + 4096); // ball idx (20480)

  auto gemm = [&](const float* A, const float* Bm, const float* bias, float* C,
                  int M, int Nn, int Kd, int lda, int ldb, int ldc,
                  long sA, long sB, long sC, int batch, int transB, int biasDim,
                  int act, float scale) {
    dim3 g(CDIV(Nn, 32), CDIV(M, 16), batch);
    k_gemm<<<g, 32, 0, stream>>>(A, Bm, bias, C, M, Nn, Kd, lda, ldb, ldc,
                                 sA, sB, sC, transB, biasDim, act, scale);
  };

  auto pai_conv = [&](const float* pts, const float* feat, const ConvP& P,
                      int N, int C, int icx, int nf, int O, float* outFeat) {
    long R = (long)BATCH * N * KNN;
    long J = (long)BATCH * N * NKERN;
    k_pk<<<1, 32, 0, stream>>>(P.A, kern, pkb);
    k_knn<<<CDIV(BATCH * N, 128), 128, 0, stream>>>(pts, idxb, BATCH, N);
    k_xf<<<CDIV((int)R, 256), 256, 0, stream>>>(pts, idxb, xfb, BATCH, N);
    k_minmax_part<<<512, 256, 0, stream>>>(xfb, R * 7, pmin, pmax);
    k_minmax_fin<<<1, 256, 0, stream>>>(pmin, pmax, 512, sb);
    for (long r0 = 0; r0 < R; r0 += CHUNKR) {
      int Rc = (int)((R - r0 < CHUNKR) ? (R - r0) : CHUNKR);
      dim3 g(CDIV(Rc, 64), 32);
      k_sincos<<<g, 64, 0, stream>>>(xfb, P.Bf, sb, bigb, r0, Rc);
      gemm(P.mlpW, bigb, P.mlpb, ffb + r0, icx, Rc, 64, 64, Rc, (int)R,
           0, 0, 0, 1, 0, 0, 0, 1.f);
    }
    k_perm<<<CDIV((int)R, 256), 256, 0, stream>>>(xfb, pkb, pmb, R);
    k_topk<<<CDIV(BATCH * N * NKERN, 256), 256, 0, stream>>>(pmb, BATCH * N);
    k_agg<<<(int)CDIV((long)BATCH * N * nf, 128), 128, 0, stream>>>(
        feat, ffb, idxb, pmb, aggb, BATCH, N, C, icx, nf);
    // conv: (O x nf) @ (nf x J) + convb, into bigb (sincos region is dead now)
    gemm(P.convW, aggb, P.convb, bigb, O, (int)J, nf, nf, (int)J, (int)J,
         0, 0, 0, 1, 0, 0, 0, 1.f);
    // residual: batched (O x C) @ (C x N)
    gemm(P.outW, feat, nullptr, resb, O, N, C, C, N, N,
         0, (long)C * N, (long)O * N, BATCH, 0, 0, 0, 1.f);
    k_combine<<<(int)CDIV((long)BATCH * O * N, 256), 256, 0, stream>>>(
        bigb, resb, outFeat, BATCH, N, O);
  };

  auto pai_pool = [&](const float* pts, const float* feat, int N, int C, int S,
                      const float* poolW, float* ptsOut, float* featOut) {
    k_fps<<<BATCH, 256, 0, stream>>>(pts, distb, fidxb, N, S);
    k_ball<<<CDIV(BATCH * S, 128), 128, 0, stream>>>(pts, fidxb, nidxb,
                                                     BATCH, N, S, 0.04f);
    k_ptsmax<<<CDIV(BATCH * S, 128), 128, 0, stream>>>(pts, nidxb, ptsOut,
                                                       BATCH, N, S);
    k_fgather<<<(int)CDIV((long)BATCH * C * S, 256), 256, 0, stream>>>(
        feat, nidxb, fgb, BATCH, N, C, S);
    gemm(poolW, fgb, nullptr, fgob, C, S * KDQ, C, C, S * KDQ, S * KDQ,
         0, (long)C * S * KDQ, (long)C * S * KDQ, BATCH, 0, 0, 0, 1.f);
    k_poolmax<<<(int)CDIV((long)BATCH * C * S, 256), 256, 0, stream>>>(
        fgob, featOut, BATCH, C, S);
  };

  // ---- network ----
  const int Ns[5]   = {2048, 512, 256, 128, 64};
  const int Cs[5]   = {3, 64, 64, 128, 256};
  const int icxs[5] = {3, 32, 32, 64, 128};
  const int nfs[5]  = {12, 192, 192, 384, 768};
  const int Os[5]   = {64, 64, 128, 256, 512};

  k_xpose<<<CDIV(BATCH * 2048 * 3, 256), 256, 0, stream>>>(x, ptsA, BATCH, 2048);

  float* ptsCur = ptsA;
  float* ptsNxt = ptsB;
  const float* featCur = x;
  int fb = 0;
  for (int L = 0; L < 5; ++L) {
    float* fout = featBufs[fb];
    pai_conv(ptsCur, featCur, cv[L], Ns[L], Cs[L], icxs[L], nfs[L], Os[L], fout);
    featCur = fout; fb ^= 1;
    if (L < 4) {
      float* fpool = featBufs[fb];
      pai_pool(ptsCur, featCur, Ns[L], Os[L], Ns[L + 1], pw[L], ptsNxt, fpool);
      featCur = fpool; fb ^= 1;
      float* t = ptsCur; ptsCur = ptsNxt; ptsNxt = t;
    }
  }

  // conv5: z = gelu((conv5W @ f) * BN_S), batched; z -> unused feat buffer
  float* zb = featBufs[fb];
  gemm(conv5W, featCur, nullptr, zb, 1024, 64, 512, 512, 64, 64,
       0, (long)512 * 64, (long)1024 * 64, BATCH, 0, 0, 1, BN_SC);
  k_head<<<CDIV(BATCH * 1024, 256), 256, 0, stream>>>(zb, hb, BATCH);

  // head linears (x @ W^T): transB
  gemm(hb,  lin1W, nullptr, h1b, 8, 512, 2048, 2048, 2048, 512,
       0, 0, 0, 1, 1, 0, 1, BN_SC);
  gemm(h1b, lin2W, lin2b,  h2b, 8, 256, 512, 512, 512, 256,
       0, 0, 0, 1, 1, 1, 1, BN_SC);
  gemm(h2b, lin3W, lin3b,  (float*)d_out, 8, 40, 256, 256, 256, 40,
       0, 0, 0, 1, 1, 1, 0, 1.f);
}